// YOLOv7_30880814858449
// MI455X (gfx1250) — compile-verified
//
#include <hip/hip_runtime.h>
#include <stdint.h>

#define BATCH   8
#define NANCH   25200
#define NDIM    85
#define NC      80
#define KTOP    4096
#define MAXDET  300
#define CONF_T  0.001f
#define IOU_T   0.6f
#define MAX_WH  4096.0f
#define NBUCKET 8192
#define CANDCAP 8192

typedef float v2f __attribute__((ext_vector_type(2)));
typedef float v8f __attribute__((ext_vector_type(8)));
typedef unsigned int u32;
typedef unsigned long long u64;
typedef unsigned short u16;

// ---------------- 0: zero scratch + output ----------------
__global__ void k_zero(u32* __restrict__ hist, u32* __restrict__ cnt,
                       float* __restrict__ out) {
    int i = blockIdx.x * 256 + threadIdx.x;
    if (i < BATCH * NBUCKET) hist[i] = 0u;
    if (i < BATCH) cnt[i] = 0u;
    if (i < BATCH * MAXDET * 6) out[i] = 0.0f;
}

// ---------------- 1: per-batch score histogram (bucket = float bits >> 17) ----
__global__ void k_hist(const float* __restrict__ pred, u32* __restrict__ hist) {
    __shared__ u32 lh[NBUCKET];
    int b = blockIdx.y;
    int a = blockIdx.x * blockDim.x + threadIdx.x;
    for (int t = threadIdx.x; t < NBUCKET; t += blockDim.x) lh[t] = 0u;
    __syncthreads();
    if (a < NANCH) {
        const float* row = pred + ((size_t)b * NANCH + a) * NDIM;
        __builtin_prefetch(row, 0, 3);   // global_prefetch_b8
        float obj = row[4];
        if (obj > CONF_T) {
            for (int c = 0; c < NC; ++c) {
                float conf = obj * row[5 + c];
                if (conf > CONF_T) {
                    u32 bu = __float_as_uint(conf) >> 17;
                    if (bu > NBUCKET - 1) bu = NBUCKET - 1;
                    atomicAdd(&lh[bu], 1u);
                }
            }
        }
    }
    __syncthreads();
    u32* gh = hist + (size_t)b * NBUCKET;
    for (int t = threadIdx.x; t < NBUCKET; t += blockDim.x)
        if (lh[t]) atomicAdd(&gh[t], lh[t]);
}

// ---------------- 2: find cutoff bucket holding the KTOP-th score ------------
__global__ void k_cutoff(const u32* __restrict__ hist, u32* __restrict__ cut) {
    int b = blockIdx.x;
    if (threadIdx.x == 0) {
        const u32* h = hist + (size_t)b * NBUCKET;
        u32 cum = 0;
        int bu = NBUCKET - 1;
        for (; bu >= 0; --bu) { cum += h[bu]; if (cum >= KTOP) break; }
        if (bu < 0) bu = 0;
        cut[b] = (u32)bu;
    }
}

// ---------------- 3: compact candidates above cutoff into packed 64b keys ----
__global__ void k_compact(const float* __restrict__ pred,
                          const u32* __restrict__ cut,
                          u32* __restrict__ cnt, u64* __restrict__ cand) {
    int b = blockIdx.y;
    int a = blockIdx.x * blockDim.x + threadIdx.x;
    if (a >= NANCH) return;
    u32 cb = cut[b];
    const float* row = pred + ((size_t)b * NANCH + a) * NDIM;
    float obj = row[4];
    if (obj <= CONF_T) return;
    for (int c = 0; c < NC; ++c) {
        float conf = obj * row[5 + c];
        if (conf > CONF_T) {
            u32 bits = __float_as_uint(conf);
            u32 bu = bits >> 17;
            if (bu > NBUCKET - 1) bu = NBUCKET - 1;
            if (bu >= cb) {
                u32 pos = atomicAdd(&cnt[b], 1u);
                if (pos < CANDCAP) {
                    u32 flat = (u32)a * NC + (u32)c;
                    cand[(size_t)b * CANDCAP + pos] =
                        ((u64)bits << 32) | (u64)(0xFFFFFFFFu - flat);
                }
            }
        }
    }
}

// ---------------- 4: bitonic sort (desc) in 64KB LDS + gather boxes ----------
__global__ __launch_bounds__(1024)
void k_sortgather(const float* __restrict__ pred, const u32* __restrict__ cnt,
                  const u64* __restrict__ cand, float* __restrict__ boxes,
                  float* __restrict__ scores, float* __restrict__ clsid) {
    __shared__ u64 sk[CANDCAP];           // 64 KB of the 320 KB WGP LDS
    int b = blockIdx.x;
    u32 n = cnt[b];
    if (n > CANDCAP) n = CANDCAP;
    const u64* cb = cand + (size_t)b * CANDCAP;
    for (u32 t = threadIdx.x; t < CANDCAP; t += 1024)
        sk[t] = (t < n) ? cb[t] : 0ull;
    __syncthreads();
    for (u32 k = 2; k <= CANDCAP; k <<= 1) {
        for (u32 j = k >> 1; j > 0; j >>= 1) {
            for (u32 idx = threadIdx.x; idx < CANDCAP; idx += 1024) {
                u32 p = idx ^ j;
                if (p > idx) {
                    u64 x = sk[idx], y = sk[p];
                    bool desc = ((idx & k) == 0);
                    if (desc ? (x < y) : (x > y)) { sk[idx] = y; sk[p] = x; }
                }
            }
            __syncthreads();
        }
    }
    for (u32 t = threadIdx.x; t < KTOP; t += 1024) {
        u64 key = sk[t];
        float sc = 0.f, cx = 0.f, cy = 0.f, cw = 0.f, ch = 0.f, cf = 0.f;
        if (key != 0ull) {
            u32 flat = 0xFFFFFFFFu - (u32)(key & 0xFFFFFFFFull);
            u32 a = flat / NC, c = flat % NC;
            const float* row = pred + ((size_t)b * NANCH + a) * NDIM;
            cx = row[0]; cy = row[1]; cw = row[2]; ch = row[3];
            sc = __uint_as_float((u32)(key >> 32));
            cf = (float)c;
        }
        float* bx = boxes + ((size_t)b * KTOP + t) * 4;
        bx[0] = cx - cw * 0.5f; bx[1] = cy - ch * 0.5f;
        bx[2] = cx + cw * 0.5f; bx[3] = cy + ch * 0.5f;
        scores[(size_t)b * KTOP + t] = sc;
        clsid[(size_t)b * KTOP + t] = cf;
    }
}

// ---------------- 5: 16x16 IoU tile -> suppression mask bits -----------------
// area[i]+area[j] for the whole tile via V_WMMA_F32_16X16X4_F32:
//   A (16x4): K0 = row areas, K1 = ones   ;  B (4x16): K0 = ones, K1 = col areas
// Row boxes are staged global->LDS with GLOBAL_LOAD_ASYNC_TO_LDS_B128
// (ASYNCcnt-tracked, no VGPR round trip).  iou>thr is evaluated divide-free as
// inter > thr*union (union > 0 for real boxes; degenerate rows compare false,
// matching the reference's NaN-compare-false).
__global__ __launch_bounds__(32)
void k_iou_mask(const float* __restrict__ boxes, const float* __restrict__ clsid,
                u16* __restrict__ mask) {
    __shared__ float rbx[16][4];
    __shared__ float rofs[16];
    int b = blockIdx.z;
    int i0 = blockIdx.y * 16, j0 = blockIdx.x * 16;
    int lane = threadIdx.x;
    int ln = lane & 15;
    const float* Bb = boxes + (size_t)b * KTOP * 4;
    const float* Cc = clsid + (size_t)b * KTOP;

    // async stage: 16 row boxes (16B each) directly into LDS
    if (lane < 16) {
        u64 ga = (u64)(const void*)(Bb + (size_t)(i0 + ln) * 4);
        u32 ldst = (u32)(uintptr_t)(&rbx[ln][0]);
        asm volatile("global_load_async_to_lds_b128 %0, %1, off"
                     :: "v"(ldst), "v"(ga) : "memory");
        rofs[ln] = Cc[i0 + ln] * MAX_WH;
    }

    // col box in registers: lane's own column n = lane & 15
    float coff = Cc[j0 + ln] * MAX_WH;
    float cx1 = Bb[(j0 + ln) * 4 + 0] + coff;
    float cy1 = Bb[(j0 + ln) * 4 + 1] + coff;
    float cx2 = Bb[(j0 + ln) * 4 + 2] + coff;
    float cy2 = Bb[(j0 + ln) * 4 + 3] + coff;
    float carea = (cx2 - cx1) * (cy2 - cy1);

    asm volatile("s_wait_asynccnt 0" ::: "memory");
    __syncthreads();

    // row area from the LDS copy (class offset cancels in the area)
    float rarea = (rbx[ln][2] - rbx[ln][0]) * (rbx[ln][3] - rbx[ln][1]);

    v2f av = {0.f, 0.f}, bv = {0.f, 0.f};
    if (lane < 16) { av.x = rarea; av.y = 1.0f; bv.x = 1.0f; bv.y = carea; }
    v8f c0 = {};
    // D[m][n] = area_row[m] + area_col[n]
    v8f d = __builtin_amdgcn_wmma_f32_16x16x4_f32(
        /*neg_a=*/false, av, /*neg_b=*/false, bv,
        /*c_mod=*/(short)0, c0, /*reuse_a=*/false, /*reuse_b=*/false);

    int hi = lane >> 4;
    u32 bals[8];
    #pragma unroll
    for (int r = 0; r < 8; ++r) {
        int mi = r + 8 * hi;                 // D lane layout: M = r (+8 for hi half)
        float ro  = rofs[mi];
        float qx1 = rbx[mi][0] + ro, qy1 = rbx[mi][1] + ro;
        float qx2 = rbx[mi][2] + ro, qy2 = rbx[mi][3] + ro;
        float ix1 = fmaxf(qx1, cx1), iy1 = fmaxf(qy1, cy1);
        float ix2 = fminf(qx2, cx2), iy2 = fminf(qy2, cy2);
        float inter = fmaxf(ix2 - ix1, 0.f) * fmaxf(iy2 - iy1, 0.f);
        int gi = i0 + mi, gj = j0 + ln;
        // iou > thr  <=>  inter > thr*(areaSum - inter), union > 0
        bool p = (inter > IOU_T * (d[r] - inter)) && (gj > gi);
        bals[r] = __builtin_amdgcn_ballot_w32(p);
    }
    // ballots are wave-uniform: lanes 0..15 each emit their own row's 16b chunk
    u32 val = 0;
    #pragma unroll
    for (int r = 0; r < 8; ++r) {
        val = (ln == r)     ? (bals[r] & 0xFFFFu) : val;
        val = (ln == r + 8) ? (bals[r] >> 16)     : val;
    }
    if (lane < 16) {
        u16* mrow = mask + (size_t)b * KTOP * 256;
        mrow[(size_t)(i0 + ln) * 256 + (j0 >> 4)] = (u16)val;
    }
}

// ---------------- 6: single-wave greedy scan + emit first 300 keeps ----------
__global__ __launch_bounds__(32)
void k_nms_scan(const float* __restrict__ boxes, const float* __restrict__ scores,
                const float* __restrict__ clsid, const u16* __restrict__ mask,
                float* __restrict__ out) {
    __shared__ u32 supp[KTOP / 32];       // 512B suppressed bitset
    int b = blockIdx.x;
    int lane = threadIdx.x;
    for (int t = lane; t < KTOP / 32; t += 32) supp[t] = 0u;
    __syncthreads();
    const float* sc = scores + (size_t)b * KTOP;
    const float* bx = boxes + (size_t)b * KTOP * 4;
    const float* cl = clsid + (size_t)b * KTOP;
    const u32* mk = (const u32*)(mask + (size_t)b * KTOP * 256);
    int cnt = 0;
    for (int i = 0; i < KTOP; ++i) {
        if (i + 1 < KTOP)   // hide L2 latency of the next mask row
            __builtin_prefetch(mk + (size_t)(i + 1) * 128 + lane * 4, 0, 3);
        bool keep = (sc[i] > CONF_T) && !((supp[i >> 5] >> (i & 31)) & 1u);
        if (keep) {
            const u32* row = mk + (size_t)i * 128;
            for (int w = 0; w < 4; ++w)
                supp[lane + w * 32] |= row[lane + w * 32];
            if (lane == 0 && cnt < MAXDET) {
                float* o = out + ((size_t)b * MAXDET + cnt) * 6;
                o[0] = bx[i * 4 + 0]; o[1] = bx[i * 4 + 1];
                o[2] = bx[i * 4 + 2]; o[3] = bx[i * 4 + 3];
                o[4] = sc[i];         o[5] = cl[i];
            }
            cnt++;
        }
        __syncthreads();
    }
}

// ---------------- host ----------------
extern "C" void kernel_launch(void* const* d_in, const int* in_sizes, int n_in,
                              void* d_out, int out_size, void* d_ws, size_t ws_size,
                              hipStream_t stream) {
    (void)in_sizes; (void)n_in; (void)out_size; (void)ws_size;
    const float* pred = (const float*)d_in[0];
    float* out = (float*)d_out;

    char* ws = (char*)d_ws;
    size_t off = 0;
    auto carve = [&](size_t bytes) -> void* {
        void* p = (void*)(ws + off);
        off = (off + bytes + 255) & ~(size_t)255;
        return p;
    };
    u32*  hist   = (u32*)carve((size_t)BATCH * NBUCKET * sizeof(u32));   // 256 KB
    u32*  cut    = (u32*)carve((size_t)BATCH * sizeof(u32));
    u32*  cnt    = (u32*)carve((size_t)BATCH * sizeof(u32));
    u64*  cand   = (u64*)carve((size_t)BATCH * CANDCAP * sizeof(u64));   // 512 KB
    float* boxes = (float*)carve((size_t)BATCH * KTOP * 4 * sizeof(float)); // 512 KB
    float* scores= (float*)carve((size_t)BATCH * KTOP * sizeof(float));  // 128 KB
    float* clsid = (float*)carve((size_t)BATCH * KTOP * sizeof(float));  // 128 KB
    u16*  mask   = (u16*)carve((size_t)BATCH * KTOP * 256 * sizeof(u16)); // 16 MB

    k_zero<<<dim3(256), dim3(256), 0, stream>>>(hist, cnt, out);

    dim3 ga((NANCH + 255) / 256, BATCH);
    k_hist<<<ga, 256, 0, stream>>>(pred, hist);
    k_cutoff<<<BATCH, 32, 0, stream>>>(hist, cut);
    k_compact<<<ga, 256, 0, stream>>>(pred, cut, cnt, cand);
    k_sortgather<<<BATCH, 1024, 0, stream>>>(pred, cnt, cand, boxes, scores, clsid);

    dim3 gm(KTOP / 16, KTOP / 16, BATCH);
    k_iou_mask<<<gm, 32, 0, stream>>>(boxes, clsid, mask);
    k_nms_scan<<<BATCH, 32, 0, stream>>>(boxes, scores, clsid, mask, out);
}